// GraphSAGEEncoder_33285996544640
// MI455X (gfx1250) — compile-verified
//
#include <hip/hip_runtime.h>
#include <math.h>

// GraphSAGE encoder for MI455X (gfx1250, wave32).
// Matrix ops use V_WMMA_F32_16X16X4_F32 (f32 A/B, f32 acc) — workload is
// memory-bound (scatter dominates), so full-precision WMMA is the right pick.
// W is staged into LDS row-pair interleaved so each B operand is one
// ds_load_b64 into an even-aligned VGPR pair (no v_dual_mov glue).

typedef __attribute__((ext_vector_type(2))) float v2f;
typedef __attribute__((ext_vector_type(8))) float v8f;

#define N_NODES 100000
#define N_EDGES 640000
#define D_IN    384
#define D_H     128
#define N_LAYERS 3

// ---------------------------------------------------------------- utilities

__global__ void zero_kernel(float* __restrict__ p, long long n) {
  long long i = (long long)blockIdx.x * blockDim.x + threadIdx.x;
  long long stride = (long long)gridDim.x * blockDim.x;
  for (; i < n; i += stride) p[i] = 0.0f;
}

__global__ void deg_kernel(const int* __restrict__ dst, float* __restrict__ deg) {
  int e = blockIdx.x * 256 + threadIdx.x;
  if (e < N_EDGES) {
    __hip_atomic_fetch_add(&deg[dst[e]], 1.0f, __ATOMIC_RELAXED,
                           __HIP_MEMORY_SCOPE_AGENT);
  }
}

__global__ void invdeg_kernel(float* __restrict__ deg) {
  int i = blockIdx.x * 256 + threadIdx.x;
  if (i < N_NODES) {
    float d = deg[i];
    deg[i] = d > 0.0f ? 1.0f / d : 0.0f;
  }
}

// qb[j] = b_in[j] + b_q[j] + sum_k q[k] * W_q[k][j]
__global__ void qproj_kernel(const float* __restrict__ q, const float* __restrict__ Wq,
                             const float* __restrict__ b_in, const float* __restrict__ b_q,
                             float* __restrict__ qb) {
  int j = threadIdx.x;  // 128 threads
  float s = b_in[j] + b_q[j];
  for (int k = 0; k < D_IN; ++k) s += q[k] * Wq[k * D_H + j];
  qb[j] = s;
}

// Stage 64 K-rows of W into LDS, row-pair interleaved:
//   sW[pr*256 + c*2 + s] = W[kc + 2*pr + s][c]   (pr = 0..31, s = 0..1)
// so the lane's B operand {W[k][c], W[k+1][c]} is one contiguous v2f.
__device__ __forceinline__ void stage_w64(float* __restrict__ sW,
                                          const float* __restrict__ Wc, int tid) {
  for (int it = tid; it < 32 * 32; it += 256) {  // 32 row-pairs x 32 col-groups
    int pr = it >> 5;
    int c0 = (it & 31) * 4;
    float4 r0 = *(const float4*)(Wc + (size_t)(2 * pr) * D_H + c0);
    float4 r1 = *(const float4*)(Wc + (size_t)(2 * pr + 1) * D_H + c0);
    v2f* dstp = (v2f*)&sW[pr * 256 + c0 * 2];
    v2f t0 = {r0.x, r1.x};
    v2f t1 = {r0.y, r1.y};
    v2f t2 = {r0.z, r1.z};
    v2f t3 = {r0.w, r1.w};
    dstp[0] = t0; dstp[1] = t1; dstp[2] = t2; dstp[3] = t3;
  }
}

// ------------------------------------------------- input GEMM: h0 = X@W + qb
// One wave computes a 16x128 strip (8 f32 16x16 tiles). K = 384.
__global__ __launch_bounds__(256)
void gemm_in_kernel(const float* __restrict__ X, const float* __restrict__ W,
                    const float* __restrict__ qb, float* __restrict__ H) {
  __shared__ float sW[64 * D_H];  // 32 KB weight staging (interleaved)
  const int tid  = threadIdx.x;
  const int wave = tid >> 5;
  const int lane = tid & 31;
  const int half = lane >> 4;
  const int ln   = lane & 15;
  const int row0 = blockIdx.x * 128 + wave * 16;
  int rowA = row0 + ln;
  if (rowA >= N_NODES) rowA = N_NODES - 1;  // clamp: EXEC stays all-1 for WMMA
  const float* Arow = X + (size_t)rowA * D_IN;

  v8f acc[8] = {};
  for (int kc = 0; kc < D_IN; kc += 64) {
    __syncthreads();
    stage_w64(sW, W + (size_t)kc * D_H, tid);
    __syncthreads();
#pragma unroll 4
    for (int kk = 0; kk < 64; kk += 4) {
      // A 16x4 layout: lane holds A[lane%16][2*half + v], v=0,1
      v2f a = *(const v2f*)(Arow + kc + kk + 2 * half);
      const v2f* brow = (const v2f*)&sW[(kk / 2 + half) * 256];
#pragma unroll
      for (int t = 0; t < 8; ++t) {
        v2f b = brow[t * 16 + ln];  // single ds_load_b64
        acc[t] = __builtin_amdgcn_wmma_f32_16x16x4_f32(
            false, a, false, b, (short)0, acc[t], false, false);
      }
    }
  }
  if (row0 < N_NODES) {
#pragma unroll
    for (int t = 0; t < 8; ++t) {
      int col = t * 16 + ln;
      float bias = qb[col];
#pragma unroll
      for (int j = 0; j < 8; ++j) {  // C/D: VGPR j -> row j + 8*half, col lane%16
        int r = row0 + j + 8 * half;
        H[(size_t)r * D_H + col] = acc[t][j] + bias;
      }
    }
  }
}

// ---------------------------------------------- edge scatter: agg[dst]+=h[src]
// Wave per edge, 4 floats per lane (coalesced 512B gather + f32 atomic adds).
__global__ __launch_bounds__(256)
void scatter_kernel(const float* __restrict__ H, const int* __restrict__ src,
                    const int* __restrict__ dst, float* __restrict__ Agg) {
  const long long gid = (long long)blockIdx.x * 256 + threadIdx.x;
  const int e = (int)(gid >> 5);
  if (e >= N_EDGES) return;
  const int c4 = (threadIdx.x & 31) * 4;
  float4 v = *(const float4*)(H + (size_t)src[e] * D_H + c4);
  float* pd = Agg + (size_t)dst[e] * D_H + c4;
  __hip_atomic_fetch_add(pd + 0, v.x, __ATOMIC_RELAXED, __HIP_MEMORY_SCOPE_AGENT);
  __hip_atomic_fetch_add(pd + 1, v.y, __ATOMIC_RELAXED, __HIP_MEMORY_SCOPE_AGENT);
  __hip_atomic_fetch_add(pd + 2, v.z, __ATOMIC_RELAXED, __HIP_MEMORY_SCOPE_AGENT);
  __hip_atomic_fetch_add(pd + 3, v.w, __ATOMIC_RELAXED, __HIP_MEMORY_SCOPE_AGENT);
}

// ---- layer: t = agg*inv_deg @ Wl + bl + h @ Wr; h' = h + LN(t); optional relu
__global__ __launch_bounds__(256)
void gemm_layer_kernel(const float* __restrict__ Hin, const float* __restrict__ Agg,
                       const float* __restrict__ invdeg,
                       const float* __restrict__ Wl, const float* __restrict__ bl,
                       const float* __restrict__ Wr,
                       const float* __restrict__ lng, const float* __restrict__ lnb,
                       float* __restrict__ Hout, int apply_relu) {
  __shared__ float smem[128 * D_H];  // 64 KB: first 8192 = W staging, then block tile
  const int tid  = threadIdx.x;
  const int wave = tid >> 5;
  const int lane = tid & 31;
  const int half = lane >> 4;
  const int ln   = lane & 15;
  const int row0 = blockIdx.x * 128 + wave * 16;
  int rowA = row0 + ln;
  if (rowA >= N_NODES) rowA = N_NODES - 1;
  const float scaleA = invdeg[rowA];

  v8f acc[8] = {};
  for (int phase = 0; phase < 2; ++phase) {
    const float* A = (phase == 0) ? Agg : Hin;
    const float* W = (phase == 0) ? Wl : Wr;
    const float  s = (phase == 0) ? scaleA : 1.0f;
    const float* Arow = A + (size_t)rowA * D_H;
    for (int kc = 0; kc < D_H; kc += 64) {
      __syncthreads();
      stage_w64(smem, W + (size_t)kc * D_H, tid);
      __syncthreads();
#pragma unroll 4
      for (int kk = 0; kk < 64; kk += 4) {
        v2f a = *(const v2f*)(Arow + kc + kk + 2 * half);
        a.x *= s; a.y *= s;
        const v2f* brow = (const v2f*)&smem[(kk / 2 + half) * 256];
#pragma unroll
        for (int t = 0; t < 8; ++t) {
          v2f b = brow[t * 16 + ln];  // single ds_load_b64
          acc[t] = __builtin_amdgcn_wmma_f32_16x16x4_f32(
              false, a, false, b, (short)0, acc[t], false, false);
        }
      }
    }
  }
  __syncthreads();
  // Spill 128x128 block tile (+ lin_l bias) to LDS for row-wise LayerNorm.
#pragma unroll
  for (int t = 0; t < 8; ++t) {
    int col = t * 16 + ln;
    float bias = bl[col];
#pragma unroll
    for (int j = 0; j < 8; ++j) {
      int lr = wave * 16 + j + 8 * half;
      smem[lr * D_H + col] = acc[t][j] + bias;
    }
  }
  __syncthreads();
  // Wave-per-row LayerNorm: 32 lanes x 4 cols, shfl_xor reductions.
  const int c4 = lane * 4;
  if (row0 < N_NODES) {
    float4 g  = *(const float4*)&lng[c4];
    float4 bb = *(const float4*)&lnb[c4];
    for (int r = 0; r < 16; ++r) {
      int lr = wave * 16 + r;
      int grow = row0 + r;
      float4 v = *(const float4*)&smem[lr * D_H + c4];
      float sum = v.x + v.y + v.z + v.w;
      float sq  = v.x * v.x + v.y * v.y + v.z * v.z + v.w * v.w;
#pragma unroll
      for (int off = 16; off >= 1; off >>= 1) {
        sum += __shfl_xor(sum, off, 32);
        sq  += __shfl_xor(sq,  off, 32);
      }
      float mu  = sum * (1.0f / D_H);
      float var = sq * (1.0f / D_H) - mu * mu;
      float rs  = rsqrtf(var + 1e-5f);
      float4 hres = *(const float4*)&Hin[(size_t)grow * D_H + c4];
      float4 o;
      o.x = hres.x + (v.x - mu) * rs * g.x + bb.x;
      o.y = hres.y + (v.y - mu) * rs * g.y + bb.y;
      o.z = hres.z + (v.z - mu) * rs * g.z + bb.z;
      o.w = hres.w + (v.w - mu) * rs * g.w + bb.w;
      if (apply_relu) {
        o.x = fmaxf(o.x, 0.0f); o.y = fmaxf(o.y, 0.0f);
        o.z = fmaxf(o.z, 0.0f); o.w = fmaxf(o.w, 0.0f);
      }
      *(float4*)&Hout[(size_t)grow * D_H + c4] = o;
    }
  }
}

// ----------------------------------------------------- attention softmax tail

__global__ __launch_bounds__(256)
void logits_kernel(const float* __restrict__ H, const float* __restrict__ watt,
                   const float* __restrict__ batt, float* __restrict__ logits,
                   float* __restrict__ bmax) {
  __shared__ float red[256];
  int i = blockIdx.x * 256 + threadIdx.x;
  float v = -1e30f;
  if (i < N_NODES) {
    float s = batt[0];
    const float* row = H + (size_t)i * D_H;
#pragma unroll 8
    for (int c = 0; c < D_H; ++c) s += row[c] * watt[c];
    logits[i] = s;
    v = s;
  }
  red[threadIdx.x] = v;
  __syncthreads();
  for (int o = 128; o >= 1; o >>= 1) {
    if (threadIdx.x < o) red[threadIdx.x] = fmaxf(red[threadIdx.x], red[threadIdx.x + o]);
    __syncthreads();
  }
  if (threadIdx.x == 0) bmax[blockIdx.x] = red[0];
}

__global__ void reduce_max_kernel(const float* __restrict__ bmax, int n,
                                  float* __restrict__ out) {
  __shared__ float red[256];
  float v = -1e30f;
  for (int i = threadIdx.x; i < n; i += 256) v = fmaxf(v, bmax[i]);
  red[threadIdx.x] = v;
  __syncthreads();
  for (int o = 128; o >= 1; o >>= 1) {
    if (threadIdx.x < o) red[threadIdx.x] = fmaxf(red[threadIdx.x], red[threadIdx.x + o]);
    __syncthreads();
  }
  if (threadIdx.x == 0) out[0] = red[0];
}

__global__ __launch_bounds__(256)
void expsum_kernel(float* __restrict__ logits, const float* __restrict__ gmax,
                   float* __restrict__ bsum) {
  __shared__ float red[256];
  int i = blockIdx.x * 256 + threadIdx.x;
  float v = 0.0f;
  if (i < N_NODES) {
    float e = expf(logits[i] - gmax[0]);
    logits[i] = e;
    v = e;
  }
  red[threadIdx.x] = v;
  __syncthreads();
  for (int o = 128; o >= 1; o >>= 1) {
    if (threadIdx.x < o) red[threadIdx.x] += red[threadIdx.x + o];
    __syncthreads();
  }
  if (threadIdx.x == 0) bsum[blockIdx.x] = red[0];
}

__global__ void reduce_sum_kernel(const float* __restrict__ bsum, int n,
                                  float* __restrict__ out) {
  __shared__ float red[256];
  float v = 0.0f;
  for (int i = threadIdx.x; i < n; i += 256) v += bsum[i];
  red[threadIdx.x] = v;
  __syncthreads();
  for (int o = 128; o >= 1; o >>= 1) {
    if (threadIdx.x < o) red[threadIdx.x] += red[threadIdx.x + o];
    __syncthreads();
  }
  if (threadIdx.x == 0) out[0] = red[0];
}

__global__ void norm_kernel(const float* __restrict__ e, const float* __restrict__ gsum,
                            float* __restrict__ out) {
  int i = blockIdx.x * 256 + threadIdx.x;
  if (i < N_NODES) out[i] = e[i] / gsum[0];
}

// --------------------------------------------------------------------- launch

extern "C" void kernel_launch(void* const* d_in, const int* in_sizes, int n_in,
                              void* d_out, int out_size, void* d_ws, size_t ws_size,
                              hipStream_t stream) {
  const float* x    = (const float*)d_in[0];
  const int*   ei   = (const int*)d_in[1];
  // d_in[2] = edge_attr: unused by the reference
  const float* q    = (const float*)d_in[3];
  const float* W_in = (const float*)d_in[4];
  const float* b_in = (const float*)d_in[5];
  const float* W_q  = (const float*)d_in[6];
  const float* b_q  = (const float*)d_in[7];
  const float* W_l  = (const float*)d_in[8];
  const float* b_l  = (const float*)d_in[9];
  const float* W_r  = (const float*)d_in[10];
  const float* ln_g = (const float*)d_in[11];
  const float* ln_b = (const float*)d_in[12];
  const float* watt = (const float*)d_in[13];
  const float* batt = (const float*)d_in[14];

  const int* srcI = ei;            // edge_index[0]
  const int* dstI = ei + N_EDGES;  // edge_index[1]

  float* hOut   = (float*)d_out;                        // h: [N_NODES, D_H]
  float* attOut = hOut + (size_t)N_NODES * D_H;         // attention: [N_NODES]

  // workspace layout (~103 MB)
  float* ws     = (float*)d_ws;
  float* hA     = ws;                                   // 12.8M floats
  float* agg    = hA + (size_t)N_NODES * D_H;           // 12.8M floats
  float* ideg   = agg + (size_t)N_NODES * D_H;          // 100k
  float* logits = ideg + N_NODES;                       // 100k
  float* qb     = logits + N_NODES;                     // 128
  float* bred   = qb + 128;                             // 4096 partials
  float* scal   = bred + 4096;                          // gmax, gsum

  const int NBLK = (N_NODES + 255) / 256;  // 391

  // in-degree -> reciprocal
  zero_kernel<<<256, 256, 0, stream>>>(ideg, N_NODES);
  deg_kernel<<<(N_EDGES + 255) / 256, 256, 0, stream>>>(dstI, ideg);
  invdeg_kernel<<<NBLK, 256, 0, stream>>>(ideg);

  // h0 = x@W_in + (b_in + q@W_q + b_q)
  qproj_kernel<<<1, 128, 0, stream>>>(q, W_q, b_in, b_q, qb);
  gemm_in_kernel<<<(N_NODES + 127) / 128, 256, 0, stream>>>(x, W_in, qb, hA);

  // layers (ping-pong so layer 2 output lands in d_out)
  const float* hcur = hA;
  float* buf[2] = { hOut, hA };
  for (int i = 0; i < N_LAYERS; ++i) {
    zero_kernel<<<1024, 256, 0, stream>>>(agg, (long long)N_NODES * D_H);
    scatter_kernel<<<(N_EDGES * 32) / 256, 256, 0, stream>>>(hcur, srcI, dstI, agg);
    float* hn = buf[i & 1];
    gemm_layer_kernel<<<(N_NODES + 127) / 128, 256, 0, stream>>>(
        hcur, agg, ideg,
        W_l + (size_t)i * D_H * D_H, b_l + i * D_H,
        W_r + (size_t)i * D_H * D_H,
        ln_g + i * D_H, ln_b + i * D_H,
        hn, (i < N_LAYERS - 1) ? 1 : 0);
    hcur = hn;
  }
  // hcur == hOut here (L0->d_out, L1->ws, L2->d_out)

  // softmax over all node logits (deterministic two-pass reductions)
  logits_kernel<<<NBLK, 256, 0, stream>>>(hcur, watt, batt, logits, bred);
  reduce_max_kernel<<<1, 256, 0, stream>>>(bred, NBLK, &scal[0]);
  expsum_kernel<<<NBLK, 256, 0, stream>>>(logits, &scal[0], bred);
  reduce_sum_kernel<<<1, 256, 0, stream>>>(bred, NBLK, &scal[1]);
  norm_kernel<<<NBLK, 256, 0, stream>>>(logits, &scal[1], attOut);
}